// Model_83270825934989
// MI455X (gfx1250) — compile-verified
//
#include <hip/hip_runtime.h>
#include <math.h>

#define B_    16
#define C_    64
#define D_    512
#define H_    8
#define SEQ   1024
#define LBL   512
#define PRED  512
#define DEC   1024
#define DFF_  2048
#define KSZ   25
#define TOPK  20

#define KC    32   // K-chunk staged in LDS per iteration (must divide all K)
#define LDSB  36   // padded LDS row stride in floats (16B-aligned rows, bank-conflict-free)

typedef __attribute__((ext_vector_type(2))) float v2f;
typedef __attribute__((ext_vector_type(8))) float v8f;

// ---------------------------------------------------------------------------
// WMMA f32 GEMM:  C[M,N] = A[M,K] * B(K,N) (+bias), optional B stored as [N,K].
// Batched via blockIdx.z with element strides sA/sB/sC.
//
// Block = 128 threads (4 waves), block tile 64(M) x 64(N):
//   wave w owns rows [mbase + 16w, +16) and all 64 N columns (4 accumulators).
// B chunk (KC x 64) is staged once per block in LDS, layout [n][k] with row
// stride LDSB=36 floats (rows 144B -> 16B aligned; 36*lid mod 64 distinct for
// lid 0..15 -> conflict-free ds_load_b64 fragment reads).
// A is streamed from global (rows are wave-private) and each A fragment feeds
// 4 consecutive WMMAs (reuse_a hint on the interior identical instructions).
//
// Fragment layout per CDNA5 ISA 7.12.2 (V_WMMA_F32_16X16X4_F32):
//   A 16x4: lane<16 -> M=lane, v0:K=0 v1:K=1 ; lane>=16 -> v0:K=2 v1:K=3
//   B  4x16: lane<16 -> N=lane, v0:K=0 v1:K=1 ; lane>=16 -> v0:K=2 v1:K=3
//   C 16x16: vgpr r, lane<16 -> M=r ; lane>=16 -> M=r+8 ; N=lane&15
// Requires: M%64==0, N%64==0, K%KC==0 (true for all GEMMs in this model).
// ---------------------------------------------------------------------------
template <int BT, int BIAS>
__global__ __launch_bounds__(128) void gemm_wmma(
    const float* __restrict__ A, const float* __restrict__ Bm,
    const float* __restrict__ bias, float* __restrict__ C,
    int M, int N, int K, long sA, long sB, long sC) {
  __shared__ float bs[64 * LDSB];
  const int wave = threadIdx.x >> 5;
  const int lane = threadIdx.x & 31;
  const int half = lane >> 4;
  const int lid = lane & 15;
  const int mbase = blockIdx.y * 64;
  const int nbase = blockIdx.x * 64;
  A += (size_t)blockIdx.z * sA;
  Bm += (size_t)blockIdx.z * sB;
  C += (size_t)blockIdx.z * sC;

  const int row = mbase + wave * 16 + lid;  // A row for this lane
  v8f acc0 = {}, acc1 = {}, acc2 = {}, acc3 = {};

  for (int k0 = 0; k0 < K; k0 += KC) {
    // ---- cooperative stage of B chunk into LDS as [n][k] ----
    if (BT) {
      // B is [N,K]: contiguous in k -> 64 rows x 8 float4, b128 LDS stores
#pragma unroll
      for (int j = 0; j < 4; ++j) {
        int tid = threadIdx.x + 128 * j;  // 0..511
        int n = tid >> 3;
        int k4 = (tid & 7) << 2;
        const float4 vv =
            *(const float4*)(Bm + (size_t)(nbase + n) * K + k0 + k4);
        float* dst = &bs[n * LDSB + k4];
        dst[0] = vv.x; dst[1] = vv.y; dst[2] = vv.z; dst[3] = vv.w;
      }
    } else {
      // B is [K,N]: coalesced 64-wide row reads, transposed scatter into LDS
#pragma unroll
      for (int j = 0; j < 16; ++j) {
        int idx = threadIdx.x + 128 * j;  // 0..2047
        int kc = idx >> 6;
        int n = idx & 63;
        bs[n * LDSB + kc] = Bm[(size_t)(k0 + kc) * N + nbase + n];
      }
    }
    __syncthreads();
    // ---- 8 WMMA k-steps over the staged chunk ----
#pragma unroll
    for (int ks = 0; ks < KC / 4; ++ks) {
      const int kk = k0 + ks * 4 + half * 2;
      v2f a;
      a.x = A[(size_t)row * K + kk];
      a.y = A[(size_t)row * K + kk + 1];
      if (ks == 0 && k0 + KC < K)
        __builtin_prefetch(&A[(size_t)row * K + kk + KC], 0, 1);
      const int lk = ks * 4 + half * 2;
      v2f b0 = *(const v2f*)&bs[(0 * 16 + lid) * LDSB + lk];
      v2f b1 = *(const v2f*)&bs[(1 * 16 + lid) * LDSB + lk];
      v2f b2 = *(const v2f*)&bs[(2 * 16 + lid) * LDSB + lk];
      v2f b3 = *(const v2f*)&bs[(3 * 16 + lid) * LDSB + lk];
      acc0 = __builtin_amdgcn_wmma_f32_16x16x4_f32(false, a, false, b0,
                                                   (short)0, acc0, false, false);
      acc1 = __builtin_amdgcn_wmma_f32_16x16x4_f32(false, a, false, b1,
                                                   (short)0, acc1, true, false);
      acc2 = __builtin_amdgcn_wmma_f32_16x16x4_f32(false, a, false, b2,
                                                   (short)0, acc2, true, false);
      acc3 = __builtin_amdgcn_wmma_f32_16x16x4_f32(false, a, false, b3,
                                                   (short)0, acc3, false, false);
    }
    __syncthreads();
  }
  // ---- store 16x64 per wave ----
  const int orow = mbase + wave * 16 + half * 8;
#pragma unroll
  for (int nt = 0; nt < 4; ++nt) {
    const int col = nbase + nt * 16 + lid;
    const float bb = BIAS ? bias[col] : 0.0f;
    const v8f& acc = nt == 0 ? acc0 : nt == 1 ? acc1 : nt == 2 ? acc2 : acc3;
#pragma unroll
    for (int r = 0; r < 8; ++r)
      C[(size_t)(orow + r) * N + col] = acc[r] + bb;
  }
}

// ---------------------------------------------------------------------------
// series_decomp: moving average (k=25, replicate pad) over time axis.
// seas = x - mm (if seas != null); mean written or accumulated (if != null).
// ---------------------------------------------------------------------------
__global__ void k_decomp(const float* __restrict__ x, float* __restrict__ seas,
                         float* __restrict__ mean, int Bn, int L, int Cc,
                         int accumMean) {
  size_t i = blockIdx.x * (size_t)blockDim.x + threadIdx.x;
  size_t total = (size_t)Bn * L * Cc;
  if (i >= total) return;
  int c = (int)(i % Cc);
  int l = (int)((i / Cc) % L);
  int b = (int)(i / ((size_t)Cc * L));
  const float* xb = x + (size_t)b * L * Cc + c;
  float s = 0.f;
#pragma unroll
  for (int w = -12; w <= 12; ++w) {
    int t = l + w;
    t = t < 0 ? 0 : (t >= L ? L - 1 : t);
    s += xb[(size_t)t * Cc];
  }
  float mm = s * (1.0f / (float)KSZ);
  if (seas) seas[i] = x[i] - mm;
  if (mean) {
    if (accumMean) mean[i] += mm; else mean[i] = mm;
  }
}

__global__ void k_add(float* __restrict__ dst, const float* __restrict__ src,
                      size_t n) {
  size_t i = blockIdx.x * (size_t)blockDim.x + threadIdx.x;
  if (i < n) dst[i] += src[i];
}

__global__ void k_gelu(float* __restrict__ x, size_t n) {
  size_t i = blockIdx.x * (size_t)blockDim.x + threadIdx.x;
  if (i < n) {
    float v = x[i];
    x[i] = 0.5f * v * (1.0f + erff(v * 0.70710678118654752f));
  }
}

// circular conv (kernel 3, pad 1 circular), optional sinusoidal pos add,
// optional accumulate into out.
__global__ void k_circconv(const float* __restrict__ x,
                           const float* __restrict__ W, float* __restrict__ out,
                           int Bn, int L, int Cin, int Cout, int addPos,
                           int accum) {
  size_t i = blockIdx.x * (size_t)blockDim.x + threadIdx.x;
  size_t total = (size_t)Bn * L * Cout;
  if (i >= total) return;
  int co = (int)(i % Cout);
  int l = (int)((i / Cout) % L);
  int b = (int)(i / ((size_t)Cout * L));
  float acc = 0.f;
  for (int k3 = 0; k3 < 3; ++k3) {
    int l2 = l + k3 - 1;
    l2 = (l2 < 0) ? l2 + L : (l2 >= L ? l2 - L : l2);
    const float* xr = x + ((size_t)b * L + l2) * Cin;
    const float* Wr = W + (size_t)k3 * Cin * Cout + co;
    for (int ci = 0; ci < Cin; ++ci) acc += xr[ci] * Wr[(size_t)ci * Cout];
  }
  if (addPos) {
    int pair = co >> 1;
    float freq = __expf((float)(2 * pair) * (-logf(10000.f) / (float)Cout));
    float ang = (float)l * freq;
    acc += (co & 1) ? __cosf(ang) : __sinf(ang);
  }
  if (accum) out[i] += acc; else out[i] = acc;
}

// meancorr[b,tau] = (1/D) sum_t gram[b, (t+tau)%L, t]
__global__ void k_meancorr(const float* __restrict__ gram,
                           float* __restrict__ out, int L) {
  int b = blockIdx.y;
  int tau = blockIdx.x;
  const float* g = gram + (size_t)b * L * L;
  float s = 0.f;
  for (int t = threadIdx.x; t < L; t += blockDim.x) {
    int r = t + tau;
    if (r >= L) r -= L;
    s += g[(size_t)r * L + t];
  }
  __shared__ float red[256];
  red[threadIdx.x] = s;
  __syncthreads();
  for (int o = 128; o > 0; o >>= 1) {
    if (threadIdx.x < o) red[threadIdx.x] += red[threadIdx.x + o];
    __syncthreads();
  }
  if (threadIdx.x == 0) out[(size_t)b * L + tau] = red[0] * (1.0f / (float)D_);
}

// per-batch top-20 lags + softmax over their correlation values
__global__ void k_topk(const float* __restrict__ mc, float* __restrict__ tc,
                       int* __restrict__ delay, int L) {
  int b = blockIdx.x;
  __shared__ float vals[1024];
  __shared__ float wsel[TOPK];
  __shared__ int isel[TOPK];
  __shared__ float rv[256];
  __shared__ int ri[256];
  for (int i = threadIdx.x; i < L; i += blockDim.x)
    vals[i] = mc[(size_t)b * L + i];
  __syncthreads();
  for (int kk = 0; kk < TOPK; ++kk) {
    float bv = -3.4e38f;
    int bi = 0;
    for (int i = threadIdx.x; i < L; i += blockDim.x)
      if (vals[i] > bv) { bv = vals[i]; bi = i; }
    rv[threadIdx.x] = bv;
    ri[threadIdx.x] = bi;
    __syncthreads();
    for (int o = 128; o > 0; o >>= 1) {
      if (threadIdx.x < o) {
        if (rv[threadIdx.x + o] > rv[threadIdx.x] ||
            (rv[threadIdx.x + o] == rv[threadIdx.x] &&
             ri[threadIdx.x + o] < ri[threadIdx.x])) {
          rv[threadIdx.x] = rv[threadIdx.x + o];
          ri[threadIdx.x] = ri[threadIdx.x + o];
        }
      }
      __syncthreads();
    }
    if (threadIdx.x == 0) {
      wsel[kk] = rv[0];
      isel[kk] = ri[0];
      vals[ri[0]] = -3.4e38f;
    }
    __syncthreads();
  }
  if (threadIdx.x == 0) {
    float m = wsel[0];
    for (int i = 1; i < TOPK; ++i) m = fmaxf(m, wsel[i]);
    float s = 0.f;
    float e[TOPK];
    for (int i = 0; i < TOPK; ++i) {
      e[i] = __expf(wsel[i] - m);
      s += e[i];
    }
    for (int i = 0; i < TOPK; ++i) {
      tc[b * TOPK + i] = e[i] / s;
      delay[b * TOPK + i] = isel[i];
    }
  }
}

// time-delay aggregation: out[b,l,d] = sum_i tc[b,i]*v[b,(l+delay[b,i])%L,d]
__global__ void k_aggregate(const float* __restrict__ v,
                            const float* __restrict__ tc,
                            const int* __restrict__ delay,
                            float* __restrict__ out, int L) {
  int b = blockIdx.y;
  __shared__ float stc[TOPK];
  __shared__ int sd[TOPK];
  if (threadIdx.x < TOPK) {
    stc[threadIdx.x] = tc[b * TOPK + threadIdx.x];
    sd[threadIdx.x] = delay[b * TOPK + threadIdx.x];
  }
  __syncthreads();
  size_t i = blockIdx.x * (size_t)blockDim.x + threadIdx.x;
  if (i >= (size_t)L * D_) return;
  int d = (int)(i % D_);
  int l = (int)(i / D_);
  const float* vb = v + (size_t)b * L * D_;
  float acc = 0.f;
#pragma unroll
  for (int kk = 0; kk < TOPK; ++kk) {
    int t = l + sd[kk];
    if (t >= L) t -= L;
    acc += stc[kk] * vb[(size_t)t * D_ + d];
  }
  out[((size_t)b * L + l) * D_ + d] = acc;
}

// LayerNorm over feature dim (block per (b,l))
__global__ void k_ln(const float* __restrict__ x, const float* __restrict__ g,
                     const float* __restrict__ be, float* __restrict__ out) {
  size_t bl = blockIdx.x;
  const float* xr = x + bl * D_;
  float s = 0.f, s2 = 0.f;
  for (int d = threadIdx.x; d < D_; d += blockDim.x) {
    float v = xr[d];
    s += v;
    s2 += v * v;
  }
  __shared__ float r1[256], r2[256];
  r1[threadIdx.x] = s;
  r2[threadIdx.x] = s2;
  __syncthreads();
  for (int o = 128; o > 0; o >>= 1) {
    if (threadIdx.x < o) {
      r1[threadIdx.x] += r1[threadIdx.x + o];
      r2[threadIdx.x] += r2[threadIdx.x + o];
    }
    __syncthreads();
  }
  float mu = r1[0] / (float)D_;
  float var = r2[0] / (float)D_ - mu * mu;
  float inv = rsqrtf(var + 1e-5f);
  for (int d = threadIdx.x; d < D_; d += blockDim.x)
    out[bl * D_ + d] = (xr[d] - mu) * inv * g[d] + be[d];
}

// subtract per-(b,d) mean over time (thread per (b,d))
__global__ void k_subtm(const float* __restrict__ x, float* __restrict__ out,
                        int L) {
  int i = blockIdx.x * blockDim.x + threadIdx.x;
  if (i >= B_ * D_) return;
  int d = i % D_;
  int b = i / D_;
  const float* xb = x + (size_t)b * L * D_ + d;
  float s = 0.f;
  for (int l = 0; l < L; ++l) s += xb[(size_t)l * D_];
  s /= (float)L;
  float* ob = out + (size_t)b * L * D_ + d;
  for (int l = 0; l < L; ++l) ob[(size_t)l * D_] = xb[(size_t)l * D_] - s;
}

// mean of x_enc over time per (b,c)
__global__ void k_tmean(const float* __restrict__ x, float* __restrict__ out,
                        int L, int Cc) {
  int i = blockIdx.x * blockDim.x + threadIdx.x;
  if (i >= B_ * Cc) return;
  int c = i % Cc;
  int b = i / Cc;
  const float* xb = x + (size_t)b * L * Cc + c;
  float s = 0.f;
  for (int l = 0; l < L; ++l) s += xb[(size_t)l * Cc];
  out[i] = s / (float)L;
}

// build decoder seasonal/trend initial sequences
__global__ void k_builddec(const float* __restrict__ seasFull,
                           const float* __restrict__ trendFull,
                           const float* __restrict__ xmean,
                           float* __restrict__ seasInit,
                           float* __restrict__ trend) {
  size_t i = blockIdx.x * (size_t)blockDim.x + threadIdx.x;
  if (i >= (size_t)B_ * DEC * C_) return;
  int c = (int)(i % C_);
  int l = (int)((i / C_) % DEC);
  int b = (int)(i / ((size_t)C_ * DEC));
  if (l < LBL) {
    size_t src = ((size_t)b * SEQ + (SEQ - LBL + l)) * C_ + c;
    seasInit[i] = seasFull[src];
    trend[i] = trendFull[src];
  } else {
    seasInit[i] = 0.f;
    trend[i] = xmean[b * C_ + c];
  }
}

__global__ void k_final(const float* __restrict__ trend,
                        const float* __restrict__ seas,
                        float* __restrict__ out) {
  size_t i = blockIdx.x * (size_t)blockDim.x + threadIdx.x;
  if (i >= (size_t)B_ * PRED * C_) return;
  int c = (int)(i % C_);
  int t = (int)((i / C_) % PRED);
  int b = (int)(i / ((size_t)C_ * PRED));
  size_t src = ((size_t)b * DEC + LBL + t) * C_ + c;
  out[i] = trend[src] + seas[src];
}

// ---------------------------------------------------------------------------
// Host orchestration
// ---------------------------------------------------------------------------
static void launch_gemm(hipStream_t st, const float* A, const float* Bm,
                        const float* bias, float* C, int M, int N, int K,
                        int batch, long sA, long sB, long sC, bool bt) {
  dim3 grid(N / 64, M / 64, batch);
  dim3 blk(128);
  if (bt) {
    if (bias)
      gemm_wmma<1, 1><<<grid, blk, 0, st>>>(A, Bm, bias, C, M, N, K, sA, sB, sC);
    else
      gemm_wmma<1, 0><<<grid, blk, 0, st>>>(A, Bm, bias, C, M, N, K, sA, sB, sC);
  } else {
    if (bias)
      gemm_wmma<0, 1><<<grid, blk, 0, st>>>(A, Bm, bias, C, M, N, K, sA, sB, sC);
    else
      gemm_wmma<0, 0><<<grid, blk, 0, st>>>(A, Bm, bias, C, M, N, K, sA, sB, sC);
  }
}

struct AcScratch {
  float *q, *k, *v, *gram, *mc, *tc;
  int* delay;
};

static void ac_layer(hipStream_t st, const float* qin, const float* kvin,
                     const float* Wq, const float* bq, const float* Wk,
                     const float* bk, const float* Wv, const float* bv,
                     const float* Wo, const float* bo, float* out,
                     const AcScratch& t, int L) {
  const int M = B_ * L;
  launch_gemm(st, qin, Wq, bq, t.q, M, D_, D_, 1, 0, 0, 0, false);
  launch_gemm(st, kvin, Wk, bk, t.k, M, D_, D_, 1, 0, 0, 0, false);
  launch_gemm(st, kvin, Wv, bv, t.v, M, D_, D_, 1, 0, 0, 0, false);
  // gram[b] = q_b @ k_b^T  -> batched, B transposed
  launch_gemm(st, t.q, t.k, nullptr, t.gram, L, L, D_, B_, (long)L * D_,
              (long)L * D_, (long)L * L, true);
  k_meancorr<<<dim3(L, B_), 256, 0, st>>>(t.gram, t.mc, L);
  k_topk<<<B_, 256, 0, st>>>(t.mc, t.tc, t.delay, L);
  // aggregation result reuses t.q (the Q projection is dead after gram)
  k_aggregate<<<dim3(((size_t)L * D_ + 255) / 256, B_), 256, 0, st>>>(
      t.v, t.tc, t.delay, t.q, L);
  launch_gemm(st, t.q, Wo, bo, out, M, D_, D_, 1, 0, 0, 0, false);
}

extern "C" void kernel_launch(void* const* d_in, const int* in_sizes, int n_in,
                              void* d_out, int out_size, void* d_ws,
                              size_t ws_size, hipStream_t stream) {
  (void)in_sizes; (void)n_in; (void)out_size; (void)ws_size;
  const float* x_enc = (const float*)d_in[0];
  // d_in[1] = x_dec: numerically unused by the reference forward pass
  const float* enc_emb_w = (const float*)d_in[2];
  const float* dec_emb_w = (const float*)d_in[3];
  const float* enc_Wq = (const float*)d_in[4];
  const float* enc_bq = (const float*)d_in[5];
  const float* dsa_Wq = (const float*)d_in[6];
  const float* dsa_bq = (const float*)d_in[7];
  const float* dca_Wq = (const float*)d_in[8];
  const float* dca_bq = (const float*)d_in[9];
  const float* enc_Wk = (const float*)d_in[10];
  const float* enc_bk = (const float*)d_in[11];
  const float* dsa_Wk = (const float*)d_in[12];
  const float* dsa_bk = (const float*)d_in[13];
  const float* dca_Wk = (const float*)d_in[14];
  const float* dca_bk = (const float*)d_in[15];
  const float* enc_Wv = (const float*)d_in[16];
  const float* enc_bv = (const float*)d_in[17];
  const float* dsa_Wv = (const float*)d_in[18];
  const float* dsa_bv = (const float*)d_in[19];
  const float* dca_Wv = (const float*)d_in[20];
  const float* dca_bv = (const float*)d_in[21];
  const float* enc_Wo = (const float*)d_in[22];
  const float* enc_bo = (const float*)d_in[23];
  const float* dsa_Wo = (const float*)d_in[24];
  const float* dsa_bo = (const float*)d_in[25];
  const float* dca_Wo = (const float*)d_in[26];
  const float* dca_bo = (const float*)d_in[27];
  const float* enc_ff1 = (const float*)d_in[28];
  const float* enc_ff2 = (const float*)d_in[29];
  const float* enc_ng = (const float*)d_in[30];
  const float* enc_nb = (const float*)d_in[31];
  const float* dec_ff1 = (const float*)d_in[32];
  const float* dec_ff2 = (const float*)d_in[33];
  const float* dec_trend_w = (const float*)d_in[34];
  const float* proj_w = (const float*)d_in[35];
  const float* proj_b = (const float*)d_in[36];
  float* out = (float*)d_out;

  const size_t NE = (size_t)B_ * SEQ * D_;  // 8,388,608 floats
  float* p = (float*)d_ws;
  float* X0 = p;   p += NE;
  float* X1 = p;   p += NE;
  float* Ab = p;   p += NE;   // attention / FFN output
  float* Qb = p;   p += NE;
  float* Kb = p;   p += NE;
  float* Vb = p;   p += NE;
  float* ENC = p;  p += NE;   // encoder output (persists through decoder)
  float* TS = p;   p += NE;   // decoder trend-sum t1+t2+t3
  float* FF = p;   p += 4 * NE;  // DFF hidden; aliases GRAM (2*NE) safely
  float* GRAM = FF;
  float* seasFull = p;  p += (size_t)B_ * SEQ * C_;
  float* trendFull = p; p += (size_t)B_ * SEQ * C_;
  float* sdec = p;      p += (size_t)B_ * DEC * C_;
  float* trendD = p;    p += (size_t)B_ * DEC * C_;
  float* xmean = p;     p += B_ * C_;
  float* mcorr = p;     p += (size_t)B_ * SEQ;
  float* tcb = p;       p += B_ * TOPK;
  int* delayb = (int*)p;

  AcScratch sc{Qb, Kb, Vb, GRAM, mcorr, tcb, delayb};

  const size_t nAct = NE;
  const size_t nFF = (size_t)B_ * SEQ * DFF_;
  const int TPB = 256;
  const int gAct = (int)((nAct + TPB - 1) / TPB);
  const int gFF = (int)((nFF + TPB - 1) / TPB);
  const int M = B_ * SEQ;  // 16384 rows (same for encoder and decoder)

  // ---------------- encoder embedding ----------------
  k_circconv<<<gAct, TPB, 0, stream>>>(x_enc, enc_emb_w, X0, B_, SEQ, C_, D_, 1, 0);

  // ---------------- encoder layers ----------------
  for (int l = 0; l < 2; ++l) {
    const size_t wo = (size_t)l * D_ * D_;
    const size_t bo = (size_t)l * D_;
    ac_layer(stream, X0, X0, enc_Wq + wo, enc_bq + bo, enc_Wk + wo, enc_bk + bo,
             enc_Wv + wo, enc_bv + bo, enc_Wo + wo, enc_bo + bo, Ab, sc, SEQ);
    k_add<<<gAct, TPB, 0, stream>>>(X0, Ab, nAct);
    k_decomp<<<gAct, TPB, 0, stream>>>(X0, X1, nullptr, B_, SEQ, D_, 0);
    // FFN
    launch_gemm(stream, X1, enc_ff1 + (size_t)l * D_ * DFF_, nullptr, FF, M,
                DFF_, D_, 1, 0, 0, 0, false);
    k_gelu<<<gFF, TPB, 0, stream>>>(FF, nFF);
    launch_gemm(stream, FF, enc_ff2 + (size_t)l * DFF_ * D_, nullptr, Ab, M, D_,
                DFF_, 1, 0, 0, 0, false);
    k_add<<<gAct, TPB, 0, stream>>>(X1, Ab, nAct);
    k_decomp<<<gAct, TPB, 0, stream>>>(X1, X0, nullptr, B_, SEQ, D_, 0);
  }
  // my_layernorm: LN over D then subtract per-(b,d) time mean
  k_ln<<<M, 256, 0, stream>>>(X0, enc_ng, enc_nb, X1);
  k_subtm<<<(B_ * D_ + 255) / 256, 256, 0, stream>>>(X1, ENC, SEQ);

  // ---------------- decoder init ----------------
  const size_t nEncRaw = (size_t)B_ * SEQ * C_;
  k_decomp<<<(int)((nEncRaw + 255) / 256), 256, 0, stream>>>(
      x_enc, seasFull, trendFull, B_, SEQ, C_, 0);
  k_tmean<<<(B_ * C_ + 255) / 256, 256, 0, stream>>>(x_enc, xmean, SEQ, C_);
  const size_t nDecRaw = (size_t)B_ * DEC * C_;
  k_builddec<<<(int)((nDecRaw + 255) / 256), 256, 0, stream>>>(
      seasFull, trendFull, xmean, sdec, trendD);
  k_circconv<<<gAct, TPB, 0, stream>>>(sdec, dec_emb_w, X0, B_, DEC, C_, D_, 1, 0);

  // ---------------- decoder layer (D_LAYERS = 1) ----------------
  {
    // self-attention
    ac_layer(stream, X0, X0, dsa_Wq, dsa_bq, dsa_Wk, dsa_bk, dsa_Wv, dsa_bv,
             dsa_Wo, dsa_bo, Ab, sc, DEC);
    k_add<<<gAct, TPB, 0, stream>>>(X0, Ab, nAct);
    k_decomp<<<gAct, TPB, 0, stream>>>(X0, X1, TS, B_, DEC, D_, 0);  // t1
    // cross-attention with encoder output
    ac_layer(stream, X1, ENC, dca_Wq, dca_bq, dca_Wk, dca_bk, dca_Wv, dca_bv,
             dca_Wo, dca_bo, Ab, sc, DEC);
    k_add<<<gAct, TPB, 0, stream>>>(X1, Ab, nAct);
    k_decomp<<<gAct, TPB, 0, stream>>>(X1, X0, TS, B_, DEC, D_, 1);  // +t2
    // FFN
    launch_gemm(stream, X0, dec_ff1, nullptr, FF, M, DFF_, D_, 1, 0, 0, 0, false);
    k_gelu<<<gFF, TPB, 0, stream>>>(FF, nFF);
    launch_gemm(stream, FF, dec_ff2, nullptr, Ab, M, D_, DFF_, 1, 0, 0, 0, false);
    k_add<<<gAct, TPB, 0, stream>>>(X0, Ab, nAct);
    k_decomp<<<gAct, TPB, 0, stream>>>(X0, X1, TS, B_, DEC, D_, 1);  // +t3
    // trend projection conv: trendD += circ_conv3(TS, dec_trend_w[0])
    k_circconv<<<(int)((nDecRaw + 255) / 256), 256, 0, stream>>>(
        TS, dec_trend_w, trendD, B_, DEC, D_, C_, 0, 1);
  }

  // ---------------- projection + output ----------------
  launch_gemm(stream, X1, proj_w, proj_b, sdec, M, C_, D_, 1, 0, 0, 0, false);
  const size_t nOut = (size_t)B_ * PRED * C_;
  k_final<<<(int)((nOut + 255) / 256), 256, 0, stream>>>(trendD, sdec, out);
}